// Resample_5463198401148
// MI455X (gfx1250) — compile-verified
//
#include <hip/hip_runtime.h>
#include <hip/hip_bf16.h>

// Problem constants (match reference).
#define BATCH 32
#define LSEQ  4096
#define DCH   256
#define NSAMP 512
#define OUTSTRIDE (NSAMP * DCH + 1)   // 131073 floats per batch row (odd!)

// Layout:
//   - each thread produces 4 consecutive output floats (one float4 of a row)
//   - 64 threads  = one output row (k) of 256 floats
//   - 256-thread block (8 wave32) = 4 consecutive output rows of one batch
//   - 128 blocks per batch  ->  b is BLOCK-uniform -> lengths[b] is a scalar load
__global__ __launch_bounds__(256) void Resample_5463198401148_kernel(
    const float* __restrict__ in,      // [B, L, D] f32
    const int*   __restrict__ lengths, // [B] i32
    float*       __restrict__ out)     // [B, N*D+1] f32
{
    const int tid = threadIdx.x;
    const int b   = blockIdx.x >> 7;                              // block-uniform
    const int k   = ((blockIdx.x & 127) << 2) | (tid >> 6);       // output frame
    const int d4  = tid & 63;                                     // float4 column

    // Scalar path: lengths[b] -> s_load; gcd(l, 512) = min(l & -l, 512).
    const int l = lengths[b];
    int g = l & (-l);
    g = (g > NSAMP) ? NSAMP : g;

    // coord = k*(l/N) + 0.5*g/N - 0.5   (N = 512 = 2^9, exact scaling)
    float coord = (float)k * ((float)l * (1.0f / NSAMP))
                + (float)g * (0.5f / NSAMP) - 0.5f;
    coord = fmaxf(coord, 0.0f);
    const int   lo   = (int)floorf(coord);
    const int   hi   = min(lo + 1, l - 1);
    const float frac = coord - (float)lo;

    // Gather two 1KB rows via coalesced b128 loads (16B-aligned: row = 256 f32).
    const float4* __restrict__ in4 = (const float4*)in;
    const long base = (long)b * (long)(LSEQ * (DCH / 4));
    const float4 vlo = in4[base + (long)lo * (DCH / 4) + d4];
    const float4 vhi = in4[base + (long)hi * (DCH / 4) + d4];

    float4 o;
    o.x = fmaf(frac, vhi.x - vlo.x, vlo.x);
    o.y = fmaf(frac, vhi.y - vlo.y, vlo.y);
    o.z = fmaf(frac, vhi.z - vlo.z, vlo.z);
    o.w = fmaf(frac, vhi.w - vlo.w, vlo.w);

    // Output batch stride is ODD (N*D+1), so b128 stores would be misaligned
    // for 3 of 4 batches; use scalar non-temporal (TH=NT) stores — streamed
    // data, keep L2 free for the reusable input rows.
    float* __restrict__ dst = out + (long)b * OUTSTRIDE + (long)k * DCH + (d4 << 2);
    __builtin_nontemporal_store(o.x, dst + 0);
    __builtin_nontemporal_store(o.y, dst + 1);
    __builtin_nontemporal_store(o.z, dst + 2);
    __builtin_nontemporal_store(o.w, dst + 3);

    // Append the length (one thread per batch).
    if (k == 0 && d4 == 0) {
        __builtin_nontemporal_store((float)l,
            out + (long)b * OUTSTRIDE + (NSAMP * DCH));
    }
}

extern "C" void kernel_launch(void* const* d_in, const int* in_sizes, int n_in,
                              void* d_out, int out_size, void* d_ws, size_t ws_size,
                              hipStream_t stream) {
    (void)in_sizes; (void)n_in; (void)d_ws; (void)ws_size; (void)out_size;
    const float* in      = (const float*)d_in[0];
    const int*   lengths = (const int*)d_in[1];
    float*       out     = (float*)d_out;

    // B*N*D/4 threads / 256 per block = 4096 blocks (128 blocks per batch).
    const int blocks = BATCH * NSAMP * DCH / 4 / 256;
    Resample_5463198401148_kernel<<<blocks, 256, 0, stream>>>(in, lengths, out);
}